// HSTUBlockPostprocessor_87763361726751
// MI455X (gfx1250) — compile-verified
//
#include <hip/hip_runtime.h>
#include <stdint.h>

// ---------------------------------------------------------------------------
// HSTU block postprocessor: jagged candidate gather + row-wise L2 normalize.
//   out[r, :] = values[src(r), :] / max(||values[src(r), :]||_2, 1e-6)
//   src(r)    = seqlen_offsets[seg+1] - num_candidates_offsets[seg+1] + r,
//               seg = searchsorted(num_candidates_offsets[1:], r, 'right')
// Memory-bound (AI ~0.25 flop/B). Strategy: per-wave async Global->LDS DMA
// (ASYNCcnt) with double buffering to hide gather latency; wave32 shuffle
// reduction for the norm; non-temporal b128 stores.
// ---------------------------------------------------------------------------

typedef __attribute__((ext_vector_type(4))) float v4f;

#define WAVES_PER_BLOCK 8
#define DIM 512                 // embedding dim (fixed by reference)
#define F4_PER_ROW (DIM / 4)    // 128 float4 per row (2048 B)
#define CHUNKS 4                // 4 x (32 lanes x 16B) = 2048 B per row

// Generic pointer into LDS: low 32 bits are the workgroup-relative LDS byte
// offset (flat LDS aperture is {SHARED_BASE,16'b0} in the high bits), which is
// exactly what GLOBAL_LOAD_ASYNC_TO_LDS wants in its VDST operand.
__device__ __forceinline__ uint32_t lds_byte_off(const void* p) {
  return (uint32_t)(uintptr_t)p;
}

__global__ __launch_bounds__(256) void cand_gather_l2norm(
    const float* __restrict__ values,
    const int*   __restrict__ so,    // seqlen_offsets          (B+1)
    const int*   __restrict__ nco,   // num_candidates_offsets  (B+1)
    float*       __restrict__ out,   // (T, DIM)
    int T, int B)
{
  __shared__ v4f smem[WAVES_PER_BLOCK * 2 * F4_PER_ROW];  // 32 KiB

  const int lane  = threadIdx.x & 31;
  const int wave  = threadIdx.x >> 5;
  const int gwave = blockIdx.x * WAVES_PER_BLOCK + wave;
  const int nwav  = gridDim.x * WAVES_PER_BLOCK;

  v4f* buf0 = &smem[(wave * 2 + 0) * F4_PER_ROW];
  v4f* buf1 = &smem[(wave * 2 + 1) * F4_PER_ROW];
  const uint32_t l0 = lds_byte_off(buf0) + (uint32_t)lane * 16u;
  const uint32_t l1 = lds_byte_off(buf1) + (uint32_t)lane * 16u;

  // Issue the 4 async b128 DMAs that stage row r into the LDS buffer at lbase.
  // Note: per CDNA5 ISA §10 async pseudocode, INST_OFFSET is added to BOTH the
  // LDS and the global address, so one immediate advances both streams.
  auto issue = [&](int r, uint32_t lbase) {
    // seg = searchsorted(nco[1:], r, 'right'); wave-uniform binary search.
    int lo = 1, hi = B;
    while (lo < hi) { int mid = (lo + hi) >> 1; if (nco[mid] <= r) lo = mid + 1; else hi = mid; }
    const uint32_t src  = (uint32_t)(so[lo] - nco[lo] + r);          // source row
    const uint32_t goff = src * (uint32_t)(DIM * 4) + (uint32_t)lane * 16u;
    asm volatile("global_load_async_to_lds_b128 %0, %1, %2 offset:0"
                 :: "v"(lbase), "v"(goff), "s"(values) : "memory");
    asm volatile("global_load_async_to_lds_b128 %0, %1, %2 offset:512"
                 :: "v"(lbase), "v"(goff), "s"(values) : "memory");
    asm volatile("global_load_async_to_lds_b128 %0, %1, %2 offset:1024"
                 :: "v"(lbase), "v"(goff), "s"(values) : "memory");
    asm volatile("global_load_async_to_lds_b128 %0, %1, %2 offset:1536"
                 :: "v"(lbase), "v"(goff), "s"(values) : "memory");
  };

  if (gwave < T) issue(gwave, l0);   // prologue prefetch into stage 0

  int stage = 0;
  for (int r = gwave; r < T; r += nwav) {
    const int rn = r + nwav;
    if (rn < T) {
      issue(rn, stage ? l0 : l1);    // prefetch next row into the other stage
      // 8 DMAs outstanding; in-order completion => oldest 4 (row r) landed.
      asm volatile("s_wait_asynccnt 4" ::: "memory");
    } else {
      asm volatile("s_wait_asynccnt 0" ::: "memory");
    }

    const v4f* buf = stage ? buf1 : buf0;
    v4f v[CHUNKS];
    float ss = 0.0f;
#pragma unroll
    for (int k = 0; k < CHUNKS; ++k) {
      v[k] = buf[k * 32 + lane];                       // ds_load_b128
      ss += v[k].x * v[k].x + v[k].y * v[k].y + v[k].z * v[k].z + v[k].w * v[k].w;
    }
    // wave32 tree reduction
#pragma unroll
    for (int m = 16; m > 0; m >>= 1) ss += __shfl_xor(ss, m, 32);

    const float sc = 1.0f / fmaxf(sqrtf(ss), 1e-6f);

    v4f* orow = (v4f*)(out + (size_t)r * DIM);
#pragma unroll
    for (int k = 0; k < CHUNKS; ++k) {
      v4f o = v[k] * sc;
      __builtin_nontemporal_store(o, &orow[k * 32 + lane]);  // TH=NT b128 store
    }
    stage ^= 1;
  }
}

// Tail outputs of the reference tuple, numerically converted into the flat
// f32 output buffer: seqlen = diff(nco) (B values) then nco itself (B+1).
__global__ void write_meta(const int* __restrict__ nco, float* __restrict__ meta, int B)
{
  const int t = blockIdx.x * blockDim.x + threadIdx.x;
  if (t < B)  meta[t] = (float)(nco[t + 1] - nco[t]);
  if (t <= B) meta[B + t] = (float)nco[t];
}

extern "C" void kernel_launch(void* const* d_in, const int* in_sizes, int n_in,
                              void* d_out, int out_size, void* d_ws, size_t ws_size,
                              hipStream_t stream) {
  const float* values = (const float*)d_in[0];
  const int*   so     = (const int*)d_in[1];
  const int*   nco    = (const int*)d_in[2];
  float*       out    = (float*)d_out;

  const int B = in_sizes[1] - 1;                    // seqlen_offsets has B+1
  // out_size = T*DIM + B (seqlen) + B+1 (offsets)
  const int T = (out_size - (2 * B + 1)) / DIM;

  if (T > 0) {
    // ~4 rows per wave so the async double-buffer pipeline has depth.
    int waves  = (T + 3) / 4;
    int blocks = (waves + WAVES_PER_BLOCK - 1) / WAVES_PER_BLOCK;
    if (blocks < 1)    blocks = 1;
    if (blocks > 4096) blocks = 4096;
    cand_gather_l2norm<<<blocks, 256, 0, stream>>>(values, so, nco, out, T, B);
  }

  const int metaThreads = B + 1;
  write_meta<<<(metaThreads + 255) / 256, 256, 0, stream>>>(
      nco, out + (size_t)T * DIM, B);
}